// PBCNN_23201413333271
// MI455X (gfx1250) — compile-verified
//
#include <hip/hip_runtime.h>
#include <stdint.h>

#define LWIN   50
#define CHUNK  256
#define BLOCK  256
#define TILE_N (CHUNK + 2 * LWIN)   // 356 complex values staged per block

typedef float v2f __attribute__((ext_vector_type(2)));
typedef float v8f __attribute__((ext_vector_type(8)));

// ---- compile-time tables -------------------------------------------------
constexpr int cnt_pairs() {
    int c = 0;
    for (int m = -25; m <= 25; ++m)
        for (int n = -25; n <= 25; ++n) {
            int p = m * n; if (p < 0) p = -p;
            if (p <= 25) ++c;
        }
    return c;
}
constexpr int NPAIR = cnt_pairs();
static_assert(NPAIR == 449, "ordered pair count mismatch");

constexpr int cnt_upairs() {
    int c = 0;
    for (int m = -25; m <= 25; ++m)
        for (int n = m; n <= 25; ++n) {
            int p = m * n; if (p < 0) p = -p;
            if (p <= 25) ++c;
        }
    return c;
}
constexpr int NU = cnt_upairs();
static_assert(NU == 230, "unordered pair count mismatch");

struct UTab {
    int offA[NU];   // byte offset of E[k-n]     relative to window center
    int offB[NU];   // byte offset of E[k-m-n]
    int offC[NU];   // byte offset of E[k-m]
    int pp[NU];     // p1 | (p2 << 16): ordered indices of (m,n) and (n,m)
};
constexpr UTab make_utab() {
    short idx[51][51] = {};
    int cnt = 0;
    for (int m = -25; m <= 25; ++m)
        for (int n = -25; n <= 25; ++n) {
            int p = m * n; if (p < 0) p = -p;
            if (p <= 25) idx[m + 25][n + 25] = (short)cnt++;
        }
    UTab t{};
    int u = 0;
    for (int m = -25; m <= 25; ++m)
        for (int n = m; n <= 25; ++n) {
            int p = m * n; if (p < 0) p = -p;
            if (p <= 25) {
                t.offA[u] = -n * 8;
                t.offB[u] = -(m + n) * 8;
                t.offC[u] = -m * 8;
                const int p1 = idx[m + 25][n + 25];
                const int p2 = idx[n + 25][m + 25];
                t.pp[u] = p1 | (p2 << 16);
                ++u;
            }
        }
    return t;
}
__constant__ UTab g_ut = make_utab();

// ---- kernel --------------------------------------------------------------
__global__ __launch_bounds__(BLOCK) void pbcnn_kernel(
    const float* __restrict__ xr, const float* __restrict__ xi,
    const float* __restrict__ task, const float* __restrict__ w1,
    const float* __restrict__ b1, const float* __restrict__ w2,
    const float* __restrict__ b2, const float* __restrict__ w3,
    const float* __restrict__ b3, float* __restrict__ out,
    int W, int nValid)
{
    __shared__ float2 sE[TILE_N];   // interleaved (re, im)
    __shared__ float4 sW1[NU];      // folded w1 rows {Re0, Re1, Im0, Im1}
    __shared__ int4   sOff[NU];     // byte offsets {-n*8, -(m+n)*8, -m*8, 0}
    __shared__ float  sSm[54];      // [0:2)=b1 [2:22)=w2 [22:32)=b2 [32:52)=w3 [52:54)=b3
    __shared__ float  sU[8 * 512];  // per-wave 32x16 layer-2 activations (D->row transpose)

    const int tid = threadIdx.x;
    const int b   = blockIdx.y;
    const int k0  = LWIN + blockIdx.x * CHUNK;

    // ---- stage E tile with CDNA5 async global->LDS copies (ASYNCcnt) ----
    {
        const uint64_t srcR = (uint64_t)(xr + (size_t)b * W);
        const uint64_t srcI = (uint64_t)(xi + (size_t)b * W);
        const unsigned ldsBase = (unsigned)(uint64_t)(void*)&sE[0]; // low 32 bits = LDS offset
        const int g0 = k0 - LWIN;                                    // >= 0 always
        for (int t = tid; t < TILE_N; t += BLOCK) {
            const int g = g0 + t;
            if (g < W) {
                unsigned lr = ldsBase + (unsigned)t * 8u;
                unsigned li = lr + 4u;
                unsigned vo = (unsigned)g * 4u;
                asm volatile("global_load_async_to_lds_b32 %0, %1, %2"
                             :: "v"(lr), "v"(vo), "s"(srcR) : "memory");
                asm volatile("global_load_async_to_lds_b32 %0, %1, %2"
                             :: "v"(li), "v"(vo), "s"(srcI) : "memory");
            }
        }
        asm volatile("s_wait_asynccnt 0x0" ::: "memory");
    }

    // ---- stage folded weights + pair offsets + tiny MLP weights ----
    for (int u = tid; u < NU; u += BLOCK) {
        sOff[u] = make_int4(g_ut.offA[u], g_ut.offB[u], g_ut.offC[u], 0);
        const int pp = g_ut.pp[u];
        const int p1 = pp & 0xffff;
        const int p2 = (pp >> 16) & 0xffff;
        float wx = w1[2 * p1],           wy = w1[2 * p1 + 1];
        float wz = w1[2 * (p1 + NPAIR)], ww = w1[2 * (p1 + NPAIR) + 1];
        if (p2 != p1) {                  // fold symmetric partner's weights
            wx += w1[2 * p2];            wy += w1[2 * p2 + 1];
            wz += w1[2 * (p2 + NPAIR)];  ww += w1[2 * (p2 + NPAIR) + 1];
        }
        sW1[u] = make_float4(wx, wy, wz, ww);
    }
    if (tid < 2)  sSm[tid]      = b1[tid];
    if (tid < 20) sSm[2 + tid]  = w2[tid];
    if (tid < 10) sSm[22 + tid] = b2[tid];
    if (tid < 20) sSm[32 + tid] = w3[tid];
    if (tid < 2)  sSm[52 + tid] = b3[tid];
    __syncthreads();

    const float scale = exp10f(0.15f * task[b * 4 + 0]);  // sqrt(10^(ti/10))^3

    // this lane's window center in LDS (byte addressed; pair offsets are bytes)
    const char* eC = (const char*)&sE[tid + LWIN];

    float h0 = 0.f, h1 = 0.f;
    #pragma unroll 4
    for (int u = 0; u < NU; ++u) {
        const int4   of = sOff[u];
        const float2 a  = *(const float2*)(eC + of.x);   // E[k-n]
        const float2 bb = *(const float2*)(eC + of.y);   // E[k-m-n]
        const float2 c  = *(const float2*)(eC + of.z);   // E[k-m]
        // t = a * conj(bb); F = t * c   (shared by (m,n) and (n,m))
        const float tr = fmaf(a.x, bb.x,  a.y * bb.y);
        const float tq = fmaf(a.y, bb.x, -a.x * bb.y);
        const float Fr = fmaf(tr, c.x, -tq * c.y);
        const float Fi = fmaf(tr, c.y,  tq * c.x);
        const float4 w = sW1[u];
        h0 = fmaf(Fr, w.x, fmaf(Fi, w.z, h0));
        h1 = fmaf(Fr, w.y, fmaf(Fi, w.w, h1));
    }
    h0 = fmaxf(fmaf(h0, scale, sSm[0]), 0.f);   // layer-1 activations
    h1 = fmaxf(fmaf(h1, scale, sSm[1]), 0.f);

    // ---- layer 2 (2 -> 10, +bias) on the matrix pipe: V_WMMA_F32_16X16X4_F32
    // A[m,k]: m = wave-local position, k = {h0,h1,0,0}. Lanes 0-15 natively
    // hold A rows 0-15 (K0 in v0, K1 in v1); lanes 16-31 must carry K2,K3 = 0.
    const int lane   = tid & 31;
    const int jcol   = lane & 15;
    const bool loHalf = lane < 16;

    const float bj0 = (jcol < 10) ? sSm[2  + jcol] : 0.f;   // w2[0, j]
    const float bj1 = (jcol < 10) ? sSm[12 + jcol] : 0.f;   // w2[1, j]
    const float cj  = (jcol < 10) ? sSm[22 + jcol] : 0.f;   // b2[j]
    v2f Bm; Bm.x = loHalf ? bj0 : 0.f;                       // K=0 row (K=2 row = 0)
            Bm.y = loHalf ? bj1 : 0.f;                       // K=1 row (K=3 row = 0)
    v8f Cm;
    #pragma unroll
    for (int r = 0; r < 8; ++r) Cm[r] = cj;                  // bias fused via C

    v2f Aa; Aa.x = loHalf ? h0 : 0.f;  Aa.y = loHalf ? h1 : 0.f;
    const float h0s = __shfl(h0, lane + 16, 32);             // positions 16-31 -> lanes 0-15
    const float h1s = __shfl(h1, lane + 16, 32);
    v2f Ab; Ab.x = loHalf ? h0s : 0.f; Ab.y = loHalf ? h1s : 0.f;

    v8f Da = __builtin_amdgcn_wmma_f32_16x16x4_f32(false, Aa, false, Bm, (short)0, Cm, false, false);
    v8f Db = __builtin_amdgcn_wmma_f32_16x16x4_f32(false, Ab, false, Bm, (short)0, Cm, false, false);

    // ReLU + transpose D layout -> per-lane rows through a per-wave LDS strip
    // (same-wave LDS ops are in-order; no barrier needed)
    float* sUw = &sU[(tid >> 5) * 512];
    float* pA  = sUw + ((lane >> 4) << 7) + jcol;            // (half*8)*16 + j
    #pragma unroll
    for (int r = 0; r < 8; ++r) {
        pA[r * 16]       = fmaxf(Da[r], 0.f);                // rows 0-15
        pA[256 + r * 16] = fmaxf(Db[r], 0.f);                // rows 16-31
    }

    // ---- layer 3 (10 -> 2) per lane from its transposed row ----
    const float* uRow = sUw + lane * 16;
    float o0 = sSm[52], o1 = sSm[53];
    #pragma unroll
    for (int j = 0; j < 10; ++j) {
        const float uj = uRow[j];
        o0 = fmaf(uj, sSm[32 + 2 * j],     o0);
        o1 = fmaf(uj, sSm[32 + 2 * j + 1], o1);
    }

    const int k = k0 + tid;
    if (k - LWIN < nValid) {
        const float2 e  = *(const float2*)eC;            // E[k]
        const size_t oi = ((size_t)b * nValid + (size_t)(k - LWIN)) * 2;
        *(float2*)(out + oi) = make_float2(o0 + e.x, o1 + e.y);
    }
}

extern "C" void kernel_launch(void* const* d_in, const int* in_sizes, int n_in,
                              void* d_out, int out_size, void* d_ws, size_t ws_size,
                              hipStream_t stream) {
    const float* xr   = (const float*)d_in[0];
    const float* xi   = (const float*)d_in[1];
    const float* task = (const float*)d_in[2];
    const float* w1   = (const float*)d_in[3];
    const float* b1   = (const float*)d_in[4];
    const float* w2   = (const float*)d_in[5];
    const float* b2   = (const float*)d_in[6];
    const float* w3   = (const float*)d_in[7];
    const float* b3   = (const float*)d_in[8];

    const int B = in_sizes[2] / 4;          // task_info is (B, 4)
    const int W = in_sizes[0] / B;          // x_real is (B, W, 1)
    const int nValid = W - 2 * LWIN;
    const int bpb = (nValid + CHUNK - 1) / CHUNK;

    dim3 grid(bpb, B);
    pbcnn_kernel<<<grid, BLOCK, 0, stream>>>(xr, xi, task, w1, b1, w2, b2, w3, b3,
                                             (float*)d_out, W, nValid);
}